// ROIAlign_3882650436218
// MI455X (gfx1250) — compile-verified
//
#include <hip/hip_runtime.h>
#include <stdint.h>

#define PH 7
#define PW 7
#define NBINS 49
#define ROI_SCALE (1.0f/16.0f)
#define BATCH 4
#define CH    256
#define CH4   64          // CH/4
#define FH    50
#define FW    50
#define HWS   (FH*FW)     // 2500
#define NROI  1000
#define GAX   14          // pooled * sampling_ratio

typedef float v4f __attribute__((ext_vector_type(4)));

// ---------------------------------------------------------------------------
// Kernel 1: NCHW -> NHWC relayout (per batch: [C, HW] -> [HW, C] transpose).
// Tiles of 32 channels x 32 spatial, staged through LDS using the CDNA5
// async global->LDS DMA path (ASYNCcnt), then written back coalesced.
// Row stride in LDS = 36 dwords (144 B): keeps b128 LDS writes 16B-aligned
// and readback ~conflict-free.
// ---------------------------------------------------------------------------
__global__ __launch_bounds__(256) void nchw_to_nhwc_async(
    const float* __restrict__ in, float* __restrict__ outT) {
  __shared__ __align__(16) float tile[32 * 36];

  const int s0 = blockIdx.x * 32;   // spatial tile origin
  const int c0 = blockIdx.y * 32;   // channel tile origin
  const int b  = blockIdx.z;
  const int t  = threadIdx.x;
  const int lane = t & 31;
  const int wave = t >> 5;

  // Phase 1: each thread issues one async b128 load (4 spatial floats of one
  // channel row) directly into LDS. 256 threads cover the full 32x32 tile.
  const int r    = wave * 4 + (lane >> 3);   // channel row within tile, 0..31
  const int sloc = (lane & 7) * 4;           // spatial offset within tile
  int gs = s0 + sloc;
  if (gs > HWS - 4) gs = HWS - 4;            // clamp; garbage slots unused
  const float* gptr = in + (((size_t)b * CH + (c0 + r)) * HWS + gs);
  unsigned ldsoff = (unsigned)(uintptr_t)(&tile[r * 36 + sloc]);
  asm volatile("global_load_async_to_lds_b128 %0, %1, off"
               :: "v"(ldsoff), "v"(gptr) : "memory");
  asm volatile("s_wait_asynccnt 0" ::: "memory");
  __syncthreads();

  // Phase 2: read transposed from LDS, store coalesced float4 along channels.
  const int i  = t >> 3;            // spatial index within tile, 0..31
  const int j4 = t & 7;             // channel float4 group, 0..7
  const int s  = s0 + i;
  if (s < HWS) {
    v4f v;
    v.x = tile[(j4 * 4 + 0) * 36 + i];
    v.y = tile[(j4 * 4 + 1) * 36 + i];
    v.z = tile[(j4 * 4 + 2) * 36 + i];
    v.w = tile[(j4 * 4 + 3) * 36 + i];
    v4f* dst = (v4f*)(outT + (((size_t)b * HWS + s) * CH + c0 + j4 * 4));
    *dst = v;   // regular-temporal: re-read by the gather kernel from L2
  }
}

// ---------------------------------------------------------------------------
// Kernel 2: one workgroup (8 wave32s) per ROI.
//  - 28 threads compute the separable y/x sample tables into LDS.
//  - Main loop: thread <- (bin, 4-channel chunk); 16 coalesced b128 gathers
//    from the NHWC map (L2 resident), vector FMA accumulate.
//  - Output tile (49 bins x 256 ch = 50 KB) is re-staged through LDS so the
//    final [C][7][7] layout goes to HBM as contiguous non-temporal b128s.
// ---------------------------------------------------------------------------
__global__ __launch_bounds__(256) void roi_align_gather(
    const float* __restrict__ fmapT, const float* __restrict__ rois,
    float* __restrict__ out) {
  __shared__ float s_wyl[GAX], s_wyh[GAX], s_wxl[GAX], s_wxh[GAX];
  __shared__ int   s_iyl[GAX], s_iyh[GAX], s_ixl[GAX], s_ixh[GAX];
  __shared__ int   s_b;
  __shared__ __align__(16) float obuf[CH * NBINS];   // 50176 B

  const int k = blockIdx.x;
  const int t = threadIdx.x;

  if (t < 2 * GAX) {
    const bool isY  = t < GAX;
    const int  g    = isY ? t : t - GAX;
    const float lim = isY ? (float)FH : (float)FW;
    const float p1  = rois[5 * k + (isY ? 2 : 1)] * ROI_SCALE;
    const float p2  = rois[5 * k + (isY ? 4 : 3)] * ROI_SCALE;
    const float rsz = fmaxf(p2 - p1, 1.0f);          // legacy aligned=False
    const float bsz = rsz * (1.0f / 7.0f);
    float c = p1 + bsz * ((float)g + 0.5f) * 0.5f;   // (g+0.5)/SR, SR=2
    const bool valid = (c > -1.0f) && (c < lim);
    c = fmaxf(c, 0.0f);
    float low = floorf(c);
    const bool top = low >= (lim - 1.0f);
    if (top) { low = lim - 1.0f; c = low; }
    const float high = top ? low : (low + 1.0f);
    const float frac = c - low;
    const float wlo  = valid ? (1.0f - frac) : 0.0f;
    const float whi  = valid ? frac : 0.0f;
    if (isY) { s_iyl[g]=(int)low; s_iyh[g]=(int)high; s_wyl[g]=wlo; s_wyh[g]=whi; }
    else     { s_ixl[g]=(int)low; s_ixh[g]=(int)high; s_wxl[g]=wlo; s_wxh[g]=whi; }
  }
  if (t == 255) s_b = (int)rois[5 * k];
  __syncthreads();

  const v4f* __restrict__ fb =
      (const v4f*)fmapT + (size_t)s_b * ((size_t)HWS * CH4);

  // 49 bins * 64 float4-chunks = 3136 items; consecutive lanes share a bin
  // and take consecutive channel chunks -> gathers are contiguous 512B runs.
  for (int w = t; w < NBINS * CH4; w += 256) {
    const int bin = w >> 6;
    const int c4  = w & 63;
    const int py  = bin / 7;
    const int px  = bin - py * 7;
    v4f acc = (v4f)0.0f;
#pragma unroll
    for (int sy = 0; sy < 2; ++sy) {
      const int   gy = 2 * py + sy;
      const int   yl = s_iyl[gy], yh = s_iyh[gy];
      const float wl = s_wyl[gy], wh = s_wyh[gy];
#pragma unroll
      for (int sx = 0; sx < 2; ++sx) {
        const int   gx = 2 * px + sx;
        const int   xl = s_ixl[gx], xh = s_ixh[gx];
        const float ul = s_wxl[gx], uh = s_wxh[gx];
        const v4f pll = fb[(size_t)(yl * FW + xl) * CH4 + c4];
        const v4f plh = fb[(size_t)(yl * FW + xh) * CH4 + c4];
        const v4f phl = fb[(size_t)(yh * FW + xl) * CH4 + c4];
        const v4f phh = fb[(size_t)(yh * FW + xh) * CH4 + c4];
        acc += (wl * ul) * pll + (wl * uh) * plh
             + (wh * ul) * phl + (wh * uh) * phh;
      }
    }
    acc *= 0.25f;   // mean over SRxSR samples
    const int cb = c4 * 4;
    obuf[(cb + 0) * NBINS + bin] = acc.x;
    obuf[(cb + 1) * NBINS + bin] = acc.y;
    obuf[(cb + 2) * NBINS + bin] = acc.z;
    obuf[(cb + 3) * NBINS + bin] = acc.w;
  }
  __syncthreads();

  // Coalesced non-temporal writeback of the [C][7][7] tile (write-once data).
  v4f* __restrict__ outk = (v4f*)(out + (size_t)k * (CH * NBINS));
  const v4f* __restrict__ ob4 = (const v4f*)obuf;
  for (int o = t; o < (CH * NBINS) / 4; o += 256) {
    __builtin_nontemporal_store(ob4[o], outk + o);
  }
}

// ---------------------------------------------------------------------------
extern "C" void kernel_launch(void* const* d_in, const int* in_sizes, int n_in,
                              void* d_out, int out_size, void* d_ws, size_t ws_size,
                              hipStream_t stream) {
  const float* input = (const float*)d_in[0];   // [B,C,H,W] fp32
  const float* rois  = (const float*)d_in[1];   // [K,5]     fp32
  float* out   = (float*)d_out;                 // [K,C,7,7] fp32
  float* fmapT = (float*)d_ws;                  // [B,H,W,C] fp32 = 10.24 MB

  dim3 tgrid((HWS + 31) / 32, CH / 32, BATCH);  // 79 x 8 x 4
  nchw_to_nhwc_async<<<tgrid, 256, 0, stream>>>(input, fmapT);
  roi_align_gather<<<NROI, 256, 0, stream>>>(fmapT, rois, out);
}